// MultiScaleDeformableAttention_2576980377957
// MI455X (gfx1250) — compile-verified
//
#include <hip/hip_runtime.h>

typedef __attribute__((ext_vector_type(2))) float v2f;
typedef __attribute__((ext_vector_type(8))) float v8f;

#define EMBED   256
#define HEADS   8
#define LEVELS  4
#define POINTS  4
#define C_DIM   32
#define BS      4
#define LEN_V   5440
#define LEN_Q   5440
#define MROWS   (BS * LEN_Q)   // 21760, == BS*LEN_V
#define KDIM    256            // K is always EMBED for every GEMM here

// ---------------------------------------------------------------------------
// WMMA fp32 GEMM: Y[M,N] = X[M,K] @ W[K,N] + bias[N], optional per-row mask.
// Block = 256 threads = 8 waves; each block owns 8 row-tiles x 1 col-tile.
// The K x 16 panel of W (16 KB) is staged into LDS once per block using the
// gfx1250 async-to-LDS DMA path (GLOBAL_LOAD_ASYNC_TO_LDS_B128, ASYNCcnt),
// then every wave streams it back with conflict-free ds_load_b32.
// Each wave accumulates its 16x16 tile with v_wmma_f32_16x16x4_f32 (exact
// fp32 accumulate, matching the reference numerics).
// ---------------------------------------------------------------------------
__global__ __launch_bounds__(256)
void wmma_gemm_bias_kernel(const float* __restrict__ X,
                           const float* __restrict__ W,
                           const float* __restrict__ bias,
                           const unsigned char* __restrict__ rowmask, // may be null
                           float* __restrict__ Y,
                           int M, int N)
{
    __shared__ float Bs[KDIM * 16];   // [k][col] panel, row stride 64B

    const int ntn  = N >> 4;                  // col tiles
    const int ct   = blockIdx.x % ntn;
    const int rb   = blockIdx.x / ntn;        // row-block of 8 tiles
    const int col0 = ct << 4;

    // ---- cooperative async copy of the B panel: KDIM rows x 16 cols -------
    {
        // KDIM*16 floats = KDIM*4 float4 chunks over 256 threads
        for (int idx = threadIdx.x; idx < KDIM * 4; idx += 256) {
            const int k  = idx >> 2;
            const int c4 = (idx & 3) << 2;
            const float* gsrc = W + (size_t)k * N + col0 + c4;
            const unsigned           ldsoff = (unsigned)(uintptr_t)(&Bs[k * 16 + c4]);
            const unsigned long long gaddr  = (unsigned long long)(uintptr_t)gsrc;
            asm volatile("global_load_async_to_lds_b128 %0, %1, off"
                         :: "v"(ldsoff), "v"(gaddr) : "memory");
        }
        asm volatile("s_wait_asynccnt 0x0" ::: "memory");
    }
    __syncthreads();

    // ---- per-wave 16x16 tile ----------------------------------------------
    const int wave = threadIdx.x >> 5;
    const int lane = threadIdx.x & 31;
    const int row0 = (rb * 8 + wave) << 4;
    if (row0 < M) {
        const int  lrow = lane & 15;
        const bool hi   = (lane >= 16);
        const int  koff = hi ? 2 : 0;

        // A: lane (l<16) holds row M=l, K pair {k,k+1}; lanes 16-31 hold {k+2,k+3}
        const float* arow = X + (size_t)(row0 + lrow) * KDIM + koff;
        const float* bls  = &Bs[koff * 16 + lrow];

        v8f acc = {};
#pragma unroll 4
        for (int k = 0; k < KDIM; k += 4) {
            v2f a = *(const v2f*)(arow + k);   // 8B aligned (k, koff even)
            v2f b;
            b.x = bls[(k + 0) * 16];
            b.y = bls[(k + 1) * 16];
            acc = __builtin_amdgcn_wmma_f32_16x16x4_f32(
                      /*neg_a=*/false, a, /*neg_b=*/false, b,
                      /*c_mod=*/(short)0, acc, /*reuse_a=*/false, /*reuse_b=*/false);
        }

        // C/D layout: VGPR r, lanes 0-15 -> (M=r, N=lane); lanes 16-31 -> (M=r+8)
        const int ocol = col0 + lrow;
        const float bv = bias[ocol];
#pragma unroll
        for (int r = 0; r < 8; ++r) {
            const int row = row0 + r + (hi ? 8 : 0);
            float val = acc[r] + bv;
            if (rowmask) val = rowmask[row] ? val : 0.0f;
            Y[(size_t)row * N + ocol] = val;
        }
    }
}

// ---------------------------------------------------------------------------
// In-place softmax over 16 attention logits per (b, q, head).
// logits layout: [M, HEADS*16], head h owns columns h*16 .. h*16+15.
// ---------------------------------------------------------------------------
__global__ __launch_bounds__(256)
void softmax16_kernel(float* __restrict__ logits, int M)
{
    const int idx = blockIdx.x * blockDim.x + threadIdx.x;
    if (idx >= M * HEADS) return;
    const int row = idx >> 3;
    const int h   = idx & 7;
    float* p = logits + (size_t)row * (HEADS * LEVELS * POINTS) + h * (LEVELS * POINTS);

    float m = p[0];
#pragma unroll
    for (int i = 1; i < 16; ++i) m = fmaxf(m, p[i]);
    float s = 0.0f;
    float e[16];
#pragma unroll
    for (int i = 0; i < 16; ++i) { e[i] = __expf(p[i] - m); s += e[i]; }
    const float inv = 1.0f / s;
#pragma unroll
    for (int i = 0; i < 16; ++i) p[i] = e[i] * inv;
}

// ---------------------------------------------------------------------------
// Deformable sampling: one wave32 per (b, q, head); lane = channel c (C=32).
// All address math is wave-uniform except the channel, so the 4 corner
// gathers are coalesced 128B loads per corner.
// ---------------------------------------------------------------------------
__global__ __launch_bounds__(256)
void msda_sample_kernel(const float* __restrict__ v,     // [BS*LEN_V, EMBED]
                        const float* __restrict__ refp,  // [BS, LEN_Q, LEVELS, 2]
                        const float* __restrict__ off,   // [BS*LEN_Q, 256]
                        const float* __restrict__ aw,    // [BS*LEN_Q, 128]
                        float* __restrict__ mid)         // [BS*LEN_Q, EMBED]
{
    const int wave = (blockIdx.x * blockDim.x + threadIdx.x) >> 5;
    const int lane = threadIdx.x & 31;
    if (wave >= MROWS * HEADS) return;

    const int h   = wave & (HEADS - 1);
    const int row = wave >> 3;            // b*LEN_Q + q
    const int b   = row / LEN_Q;

    const int lw[LEVELS]     = {64, 32, 16, 8};
    const int lh[LEVELS]     = {64, 32, 16, 8};
    const int lstart[LEVELS] = {0, 4096, 5120, 5376};

    const float* offrow = off + (size_t)row * (HEADS * LEVELS * POINTS * 2)
                              + h * (LEVELS * POINTS * 2);
    const float* awrow  = aw + (size_t)row * (HEADS * LEVELS * POINTS)
                              + h * (LEVELS * POINTS);
    const float* refrow = refp + (size_t)row * (LEVELS * 2);

    const int ch = h * C_DIM + lane;
    const float* vb = v + (size_t)b * LEN_V * EMBED + ch;

    float acc = 0.0f;
#pragma unroll
    for (int l = 0; l < LEVELS; ++l) {
        const int   w  = lw[l], hgt = lh[l];
        const float fw = (float)w, fh = (float)hgt;
        const float rx = refrow[l * 2 + 0];
        const float ry = refrow[l * 2 + 1];
        const float* vl = vb + (size_t)lstart[l] * EMBED;
#pragma unroll
        for (int p = 0; p < POINTS; ++p) {
            const float ox = offrow[(l * POINTS + p) * 2 + 0];
            const float oy = offrow[(l * POINTS + p) * 2 + 1];
            // loc = ref + off/(w,h); x = loc.x*w - 0.5 == rx*w + ox - 0.5
            const float x = rx * fw + ox - 0.5f;
            const float y = ry * fh + oy - 0.5f;
            const float fx0 = floorf(x), fy0 = floorf(y);
            const int x0 = (int)fx0, y0 = (int)fy0;
            const int x1 = x0 + 1,   y1 = y0 + 1;
            const float tx = x - fx0, ty = y - fy0;
            const float wa = (1.0f - tx) * (1.0f - ty);  // (y0,x0)
            const float wb = (1.0f - tx) * ty;           // (y1,x0)
            const float wc = tx * (1.0f - ty);           // (y0,x1)
            const float wd = tx * ty;                    // (y1,x1)
            const bool x0in = (x0 >= 0) & (x0 < w);
            const bool x1in = (x1 >= 0) & (x1 < w);
            const bool y0in = (y0 >= 0) & (y0 < hgt);
            const bool y1in = (y1 >= 0) & (y1 < hgt);
            float s = 0.0f;
            if (x0in && y0in) s += wa * vl[(size_t)(y0 * w + x0) * EMBED];
            if (x0in && y1in) s += wb * vl[(size_t)(y1 * w + x0) * EMBED];
            if (x1in && y0in) s += wc * vl[(size_t)(y0 * w + x1) * EMBED];
            if (x1in && y1in) s += wd * vl[(size_t)(y1 * w + x1) * EMBED];
            acc += awrow[l * POINTS + p] * s;
        }
    }
    mid[(size_t)row * EMBED + ch] = acc;
}

// ---------------------------------------------------------------------------
extern "C" void kernel_launch(void* const* d_in, const int* in_sizes, int n_in,
                              void* d_out, int out_size, void* d_ws, size_t ws_size,
                              hipStream_t stream)
{
    const float*         query    = (const float*)d_in[0];
    const float*         refp     = (const float*)d_in[1];
    const float*         value    = (const float*)d_in[2];
    const unsigned char* pad_mask = (const unsigned char*)d_in[3];
    // d_in[4] = train (unused)
    const float* vproj_w = (const float*)d_in[5];
    const float* vproj_b = (const float*)d_in[6];
    const float* off_w   = (const float*)d_in[7];
    const float* off_b   = (const float*)d_in[8];
    const float* attn_w  = (const float*)d_in[9];
    const float* attn_b  = (const float*)d_in[10];
    const float* out_w   = (const float*)d_in[11];
    const float* out_b   = (const float*)d_in[12];
    float* out = (float*)d_out;

    // Workspace carve-up (floats): v | off | logits(->aw in place) | mid
    float* ws     = (float*)d_ws;
    float* v      = ws;
    float* offbuf = v      + (size_t)MROWS * EMBED;
    float* logits = offbuf + (size_t)MROWS * EMBED;
    float* mid    = logits + (size_t)MROWS * (HEADS * LEVELS * POINTS);

    const int M = MROWS;
    const int threads = 256;
    const int rowBlocks = M / 128;   // 8 row-tiles per block, M=21760 -> 170 exactly

    // 1) v = mask(value @ vproj_w + vproj_b)   [M,256]
    wmma_gemm_bias_kernel<<<rowBlocks * (EMBED / 16), threads, 0, stream>>>(
        value, vproj_w, vproj_b, pad_mask, v, M, EMBED);

    // 2) off = query @ off_w + off_b           [M,256]
    wmma_gemm_bias_kernel<<<rowBlocks * (EMBED / 16), threads, 0, stream>>>(
        query, off_w, off_b, nullptr, offbuf, M, EMBED);

    // 3) logits = query @ attn_w + attn_b      [M,128]
    wmma_gemm_bias_kernel<<<rowBlocks * (HEADS * LEVELS * POINTS / 16), threads, 0, stream>>>(
        query, attn_w, attn_b, nullptr, logits, M, HEADS * LEVELS * POINTS);

    // 4) softmax over groups of 16, in place
    {
        int n = M * HEADS;
        softmax16_kernel<<<(n + threads - 1) / threads, threads, 0, stream>>>(logits, M);
    }

    // 5) deformable bilinear sampling -> mid   [M,256]
    {
        int waves = M * HEADS;
        msda_sample_kernel<<<(waves * 32 + threads - 1) / threads, threads, 0, stream>>>(
            v, refp, offbuf, logits, mid);
    }

    // 6) out = mid @ out_w + out_b             [M,256]
    wmma_gemm_bias_kernel<<<rowBlocks * (EMBED / 16), threads, 0, stream>>>(
        mid, out_w, out_b, nullptr, out, M, EMBED);
}